// SharedStateBlock_25288767439157
// MI455X (gfx1250) — compile-verified
//
#include <hip/hip_runtime.h>
#include <math.h>

#define D_MODEL   768
#define D_INNER   1536
#define N_HEADS   24
#define HEADDIM   64
#define D_STATE   64
#define D_MLP     3072
#define K_CONV    4
#define CONV_DIM  (D_INNER + 2 * D_STATE)               // 1664
#define NPROJ     (2 * D_INNER + 2 * D_STATE + N_HEADS) // 3224
#define BATCH     2
#define SEQ       1024
#define TOKENS    (BATCH * SEQ)                         // 2048
#define EPS_RMS   1e-6f

typedef __attribute__((ext_vector_type(2))) float v2f;
typedef __attribute__((ext_vector_type(8))) float v8f;

__device__ __forceinline__ float silu_f(float x)     { return x / (1.0f + __expf(-x)); }
__device__ __forceinline__ float softplus_f(float x) { return (x > 20.0f) ? x : log1pf(__expf(x)); }

// ---------------------------------------------------------------------------
// RMSNorm: one block (256 threads) per row of D_MODEL
// ---------------------------------------------------------------------------
__global__ void rmsnorm_kernel(const float* __restrict__ x, const float* __restrict__ w,
                               float* __restrict__ o) {
  const int row = blockIdx.x;
  const float* xr = x + (size_t)row * D_MODEL;
  float* orow     = o + (size_t)row * D_MODEL;
  __shared__ float red[8];
  float s = 0.0f;
  for (int i = threadIdx.x; i < D_MODEL; i += 256) { float v = xr[i]; s += v * v; }
  for (int off = 16; off > 0; off >>= 1) s += __shfl_down(s, off, 32);
  const int wave = threadIdx.x >> 5, lane = threadIdx.x & 31;
  if (lane == 0) red[wave] = s;
  __syncthreads();
  if (threadIdx.x == 0) {
    float t = 0.0f;
    for (int i = 0; i < 8; ++i) t += red[i];
    red[0] = rsqrtf(t / (float)D_MODEL + EPS_RMS);
  }
  __syncthreads();
  const float r = red[0];
  for (int i = threadIdx.x; i < D_MODEL; i += 256) orow[i] = xr[i] * r * w[i];
}

// ---------------------------------------------------------------------------
// fp32 WMMA GEMM: C[M,N] = A[M,K] @ W[K,N] (row-major).
// Each wave owns TWO adjacent 16x16 tiles sharing one A fragment
// (2x v_wmma_f32_16x16x4_f32 per K-step). Block = 8 waves = 256 cols.
// EPI==1: add residual RES before store.
// ---------------------------------------------------------------------------
template <int EPI>
__global__ void gemm_f32_wmma(const float* __restrict__ A, const float* __restrict__ W,
                              float* __restrict__ C, const float* __restrict__ RES,
                              int N, int K) {
  const int wave = threadIdx.x >> 5;
  const int lane = threadIdx.x & 31;
  const int half = lane >> 4;
  const int lm   = lane & 15;
  const int col0 = (blockIdx.x * 8 + wave) * 32 + lm;  // first tile
  const int col1 = col0 + 16;                          // second tile
  const int row0 = blockIdx.y * 16;
  const int c0c  = (col0 < N) ? col0 : (N - 1);        // clamp OOB loads
  const int c1c  = (col1 < N) ? col1 : (N - 1);
  const float* __restrict__ Arow = A + (size_t)(row0 + lm) * K;

  v8f acc0 = {}, acc1 = {};
#pragma unroll 4
  for (int k0 = 0; k0 < K; k0 += 4) {
    const int ka = k0 + 2 * half;
    const float* __restrict__ Wk0 = W + (size_t)ka * N;
    const float* __restrict__ Wk1 = Wk0 + N;
    v2f a, b0, b1;
    a.x  = Arow[ka];
    a.y  = Arow[ka + 1];
    b0.x = Wk0[c0c];  b0.y = Wk1[c0c];
    b1.x = Wk0[c1c];  b1.y = Wk1[c1c];
    acc0 = __builtin_amdgcn_wmma_f32_16x16x4_f32(false, a, false, b0, (short)0, acc0,
                                                 false, false);
    acc1 = __builtin_amdgcn_wmma_f32_16x16x4_f32(false, a, false, b1, (short)0, acc1,
                                                 false, false);
  }
#pragma unroll
  for (int r = 0; r < 8; ++r) {
    const int row = row0 + half * 8 + r;
    if (col0 < N) {
      float v = acc0[r];
      if (EPI == 1) v += RES[(size_t)row * N + col0];
      C[(size_t)row * N + col0] = v;
    }
    if (col1 < N) {
      float v = acc1[r];
      if (EPI == 1) v += RES[(size_t)row * N + col1];
      C[(size_t)row * N + col1] = v;
    }
  }
}

// ---------------------------------------------------------------------------
// Fused SwiGLU gate/up GEMM: Out = silu(A@Wg) * (A@Wu).
// Two column tiles per wave, one shared A fragment -> 4 WMMAs per K-step.
// ---------------------------------------------------------------------------
__global__ void gateup_f32_wmma(const float* __restrict__ A, const float* __restrict__ Wg,
                                const float* __restrict__ Wu, float* __restrict__ Out,
                                int N, int K) {
  const int wave = threadIdx.x >> 5;
  const int lane = threadIdx.x & 31;
  const int half = lane >> 4;
  const int lm   = lane & 15;
  const int col0 = (blockIdx.x * 8 + wave) * 32 + lm;
  const int col1 = col0 + 16;
  const int row0 = blockIdx.y * 16;
  const int c0c  = (col0 < N) ? col0 : (N - 1);
  const int c1c  = (col1 < N) ? col1 : (N - 1);
  const float* __restrict__ Arow = A + (size_t)(row0 + lm) * K;

  v8f ag0 = {}, ag1 = {}, au0 = {}, au1 = {};
#pragma unroll 2
  for (int k0 = 0; k0 < K; k0 += 4) {
    const int ka = k0 + 2 * half;
    const float* __restrict__ Wg0 = Wg + (size_t)ka * N;
    const float* __restrict__ Wg1 = Wg0 + N;
    const float* __restrict__ Wu0 = Wu + (size_t)ka * N;
    const float* __restrict__ Wu1 = Wu0 + N;
    v2f a, bg0, bg1, bu0, bu1;
    a.x   = Arow[ka];
    a.y   = Arow[ka + 1];
    bg0.x = Wg0[c0c];  bg0.y = Wg1[c0c];
    bg1.x = Wg0[c1c];  bg1.y = Wg1[c1c];
    bu0.x = Wu0[c0c];  bu0.y = Wu1[c0c];
    bu1.x = Wu0[c1c];  bu1.y = Wu1[c1c];
    ag0 = __builtin_amdgcn_wmma_f32_16x16x4_f32(false, a, false, bg0, (short)0, ag0,
                                                false, false);
    ag1 = __builtin_amdgcn_wmma_f32_16x16x4_f32(false, a, false, bg1, (short)0, ag1,
                                                false, false);
    au0 = __builtin_amdgcn_wmma_f32_16x16x4_f32(false, a, false, bu0, (short)0, au0,
                                                false, false);
    au1 = __builtin_amdgcn_wmma_f32_16x16x4_f32(false, a, false, bu1, (short)0, au1,
                                                false, false);
  }
#pragma unroll
  for (int r = 0; r < 8; ++r) {
    const int row = row0 + half * 8 + r;
    if (col0 < N) Out[(size_t)row * N + col0] = silu_f(ag0[r]) * au0[r];
    if (col1 < N) Out[(size_t)row * N + col1] = silu_f(ag1[r]) * au1[r];
  }
}

// ---------------------------------------------------------------------------
// Depthwise causal conv (K=4) over xBC slice of zxbcdt, + bias, + SiLU
// ---------------------------------------------------------------------------
__global__ void conv_silu_kernel(const float* __restrict__ zx, const float* __restrict__ cw,
                                 const float* __restrict__ cb, float* __restrict__ out) {
  const int idx = blockIdx.x * blockDim.x + threadIdx.x;
  if (idx >= TOKENS * CONV_DIM) return;
  const int c   = idx % CONV_DIM;
  const int tok = idx / CONV_DIM;
  const int b   = tok / SEQ, l = tok % SEQ;
  const float* base = zx + (size_t)b * SEQ * NPROJ + D_INNER + c;
  float acc = cb[c];
#pragma unroll
  for (int j = 0; j < K_CONV; ++j) {
    const int ll = l - (K_CONV - 1) + j;
    if (ll >= 0) acc += cw[c * K_CONV + j] * base[(size_t)ll * NPROJ];
  }
  out[(size_t)tok * CONV_DIM + c] = silu_f(acc);
}

// ---------------------------------------------------------------------------
// dt = softplus(zxbcdt[..., -N_HEADS:] + dt_bias)
// ---------------------------------------------------------------------------
__global__ void dt_kernel(const float* __restrict__ zx, const float* __restrict__ dt_bias,
                          float* __restrict__ dtsp) {
  const int idx = blockIdx.x * blockDim.x + threadIdx.x;
  if (idx >= TOKENS * N_HEADS) return;
  const int h   = idx % N_HEADS;
  const int tok = idx / N_HEADS;
  dtsp[idx] = softplus_f(zx[(size_t)tok * NPROJ + (2 * D_INNER + 2 * D_STATE) + h] +
                         dt_bias[h]);
}

// ---------------------------------------------------------------------------
// SSM scan, software-pipelined: one block per (batch, head); lane p keeps the
// 64-wide state in registers. Double-buffered LDS for B/C and register
// prefetch of next-step x/dt/z -> exactly ONE barrier per timestep and no
// load latency on the recurrence critical path. Fuses the D skip-connection
// and silu(z) gating into the epilogue.
// ---------------------------------------------------------------------------
__global__ void ssm_scan_kernel(const float* __restrict__ xBC, const float* __restrict__ dtsp,
                                const float* __restrict__ A_log, const float* __restrict__ Dp,
                                const float* __restrict__ zx, float* __restrict__ yz) {
  const int b  = blockIdx.x / N_HEADS;
  const int hd = blockIdx.x % N_HEADS;
  const int p  = threadIdx.x;  // 0..63
  const float A  = -__expf(A_log[hd]);
  const float Dh = Dp[hd];
  float hreg[D_STATE];
#pragma unroll
  for (int n = 0; n < D_STATE; ++n) hreg[n] = 0.0f;
  __shared__ float Bsh[2][D_STATE], Csh[2][D_STATE];
  const float* xb = xBC + (size_t)b * SEQ * CONV_DIM;
  const float* zb = zx  + (size_t)b * SEQ * NPROJ;
  float* yb       = yz  + (size_t)b * SEQ * D_INNER;

  // prologue: stage t=0
  Bsh[0][p] = xb[D_INNER + p];
  Csh[0][p] = xb[D_INNER + D_STATE + p];
  float xv  = xb[hd * HEADDIM + p];
  float dtv = dtsp[((size_t)(b * SEQ)) * N_HEADS + hd];
  float zv  = zb[hd * HEADDIM + p];
  __syncthreads();

  for (int t = 0; t < SEQ; ++t) {
    const int cur = t & 1, nxt = cur ^ 1;
    float xv_n = 0.0f, dtv_n = 0.0f, zv_n = 0.0f;
    if (t + 1 < SEQ) {  // prefetch t+1 while computing t
      const float* xrn = xb + (size_t)(t + 1) * CONV_DIM;
      Bsh[nxt][p] = xrn[D_INNER + p];
      Csh[nxt][p] = xrn[D_INNER + D_STATE + p];
      xv_n  = xrn[hd * HEADDIM + p];
      dtv_n = dtsp[((size_t)(b * SEQ + t + 1)) * N_HEADS + hd];
      zv_n  = zb[(size_t)(t + 1) * NPROJ + hd * HEADDIM + p];
    }
    const float dec = __expf(dtv * A);
    const float dtx = dtv * xv;
    float y = 0.0f;
#pragma unroll
    for (int n = 0; n < D_STATE; ++n) {
      hreg[n] = dec * hreg[n] + dtx * Bsh[cur][n];
      y += hreg[n] * Csh[cur][n];
    }
    y += Dh * xv;
    yb[(size_t)t * D_INNER + hd * HEADDIM + p] = y * silu_f(zv);
    xv = xv_n; dtv = dtv_n; zv = zv_n;
    __syncthreads();  // next buffer staged; cur buffer fully consumed
  }
}

// ---------------------------------------------------------------------------
extern "C" void kernel_launch(void* const* d_in, const int* in_sizes, int n_in,
                              void* d_out, int out_size, void* d_ws, size_t ws_size,
                              hipStream_t stream) {
  const float* x        = (const float*)d_in[0];
  const float* norm1_w  = (const float*)d_in[1];
  const float* in_proj  = (const float*)d_in[2];
  const float* conv_w   = (const float*)d_in[3];
  const float* conv_b   = (const float*)d_in[4];
  const float* dt_bias  = (const float*)d_in[5];
  const float* A_log    = (const float*)d_in[6];
  const float* Dp       = (const float*)d_in[7];
  const float* out_proj = (const float*)d_in[8];
  const float* norm2_w  = (const float*)d_in[9];
  const float* w_gate   = (const float*)d_in[10];
  const float* w_up     = (const float*)d_in[11];
  const float* w_down   = (const float*)d_in[12];
  float* out = (float*)d_out;

  // workspace layout (floats)
  float* ws    = (float*)d_ws;
  float* u     = ws;                               // TOKENS*D_MODEL
  float* zxb   = u     + (size_t)TOKENS * D_MODEL; // TOKENS*NPROJ
  float* cx    = zxb   + (size_t)TOKENS * NPROJ;   // TOKENS*CONV_DIM
  float* dtsp  = cx    + (size_t)TOKENS * CONV_DIM;// TOKENS*N_HEADS
  float* yzbuf = dtsp  + (size_t)TOKENS * N_HEADS; // TOKENS*D_INNER
  float* hbuf  = yzbuf + (size_t)TOKENS * D_INNER; // TOKENS*D_MODEL
  float* vbuf  = hbuf  + (size_t)TOKENS * D_MODEL; // TOKENS*D_MODEL
  float* mact  = vbuf  + (size_t)TOKENS * D_MODEL; // TOKENS*D_MLP

  // 1) u = rmsnorm(x, norm1_w)
  rmsnorm_kernel<<<TOKENS, 256, 0, stream>>>(x, norm1_w, u);

  // 2) zxbcdt = u @ in_proj   [2048 x 3224], K=768
  {
    dim3 grid((NPROJ + 255) / 256, TOKENS / 16);
    gemm_f32_wmma<0><<<grid, 256, 0, stream>>>(u, in_proj, zxb, nullptr, NPROJ, D_MODEL);
  }

  // 3) depthwise conv + SiLU on xBC slice
  conv_silu_kernel<<<(TOKENS * CONV_DIM + 255) / 256, 256, 0, stream>>>(zxb, conv_w,
                                                                        conv_b, cx);

  // 4) dt = softplus(... + dt_bias)
  dt_kernel<<<(TOKENS * N_HEADS + 255) / 256, 256, 0, stream>>>(zxb, dt_bias, dtsp);

  // 5) pipelined SSM scan + D skip + silu(z) gating -> yz [2048 x 1536]
  ssm_scan_kernel<<<BATCH * N_HEADS, HEADDIM, 0, stream>>>(cx, dtsp, A_log, Dp, zxb,
                                                           yzbuf);

  // 6) h = yz @ out_proj   [2048 x 768], K=1536
  {
    dim3 grid((D_MODEL + 255) / 256, TOKENS / 16);
    gemm_f32_wmma<0><<<grid, 256, 0, stream>>>(yzbuf, out_proj, hbuf, nullptr, D_MODEL,
                                               D_INNER);
  }

  // 7) v = rmsnorm(h, norm2_w)
  rmsnorm_kernel<<<TOKENS, 256, 0, stream>>>(hbuf, norm2_w, vbuf);

  // 8) m_act = silu(v@w_gate) * (v@w_up)   [2048 x 3072], K=768
  {
    dim3 grid((D_MLP + 255) / 256, TOKENS / 16);
    gateup_f32_wmma<<<grid, 256, 0, stream>>>(vbuf, w_gate, w_up, mact, D_MLP, D_MODEL);
  }

  // 9) out = m_act @ w_down + h   [2048 x 768], K=3072
  {
    dim3 grid((D_MODEL + 255) / 256, TOKENS / 16);
    gemm_f32_wmma<1><<<grid, 256, 0, stream>>>(mact, w_down, out, hbuf, D_MODEL, D_MLP);
  }
}